// GCN_74577812128213
// MI455X (gfx1250) — compile-verified
//
#include <hip/hip_runtime.h>
#include <stddef.h>

// ---------------------------------------------------------------------------
// GCN 2-layer forward for MI455X (gfx1250, wave32).
//   h   = relu( Agg(x @ W1) + b1 )
//   out =       Agg(h @ W2) + b2
// where Agg(y)[i] = sum_{(j->i) in E} dis[j]*dis[i]*y[j] + dis[i]^2*y[i]
// and dis = rsqrt(1 + indeg).
//
// GEMMs use V_WMMA_F32_16X16X4_F32 (f32 in/out, exact-precision path); B
// fragments come from a transposed, pad-66 LDS image of W so each fragment is
// one aligned ds_load_b64 (no VGPR packing moves between WMMAs).
// Edge aggregation is the bandwidth-dominant phase: one wave per edge,
// coalesced 256B gather + 64 global_atomic_add_f32 per edge, all L2-resident
// (working set ~77MB << 192MB L2).
// ---------------------------------------------------------------------------

typedef float v2f __attribute__((ext_vector_type(2)));
typedef float v8f __attribute__((ext_vector_type(8)));

#define FDIM 64
#define WT_STRIDE 66   // pad: keeps b64 fragments 8B-aligned, banks ~conflict-free

// ---------------- elementwise helpers ----------------

__global__ void gcn_fill_f32(float* __restrict__ p, float v, int n) {
  int i = blockIdx.x * blockDim.x + threadIdx.x;
  if (i < n) p[i] = v;
}

__global__ void gcn_degree_accum(const int* __restrict__ dst,
                                 float* __restrict__ deg, int E) {
  int stride = gridDim.x * blockDim.x;
  for (int e = blockIdx.x * blockDim.x + threadIdx.x; e < E; e += stride)
    atomicAdd(&deg[dst[e]], 1.0f);
}

__global__ void gcn_rsqrt_inplace(float* __restrict__ d, int n) {
  int i = blockIdx.x * blockDim.x + threadIdx.x;
  if (i < n) d[i] = rsqrtf(d[i]);
}

// agg[i,f] = (relu?)( agg[i,f] + dis[i]^2 * xw[i,f] + bias[f] )
__global__ void gcn_finalize(float* __restrict__ agg,
                             const float* __restrict__ xw,
                             const float* __restrict__ dis,
                             const float* __restrict__ bias,
                             int total, int do_relu) {
  int i = blockIdx.x * blockDim.x + threadIdx.x;
  if (i >= total) return;
  int node = i >> 6;        // FDIM == 64
  int f    = i & (FDIM - 1);
  float di = dis[node];
  float v  = agg[i] + di * di * xw[i] + bias[f];
  if (do_relu) v = fmaxf(v, 0.0f);
  agg[i] = v;
}

// ---------------- WMMA GEMM: Y[N,64] = X[N,64] @ W[64,64] ----------------
// One wave computes a 16x64 output tile via 4 accumulators of
// V_WMMA_F32_16X16X4_F32, walking K in steps of 4. W is staged TRANSPOSED in
// LDS (Wt[n][k], row stride 66) so B fragments are single ds_load_b64s.
//
// Fragment layouts per CDNA5 ISA 7.12.2 (wave32):
//   A (16x4):  lanes 0-15 -> M=lane,    v[0]=K0, v[1]=K1
//              lanes16-31 -> M=lane-16, v[0]=K2, v[1]=K3
//   B (4x16):  lanes 0-15 -> N=lane,    v[0]=K0 row, v[1]=K1 row
//              lanes16-31 -> N=lane-16, v[0]=K2 row, v[1]=K3 row
//   C/D(16x16): VGPR r holds M=r (lanes 0-15, N=lane) and M=r+8 (lanes 16-31)
__global__ void gcn_gemm_wmma(const float* __restrict__ X,
                              const float* __restrict__ W,
                              float* __restrict__ Y, int nrows) {
  __shared__ float Wt[FDIM * WT_STRIDE];   // ~16.9 KB, Wt[n*66 + k] = W[k*64 + n]
  int t = threadIdx.x;
  for (int i = t; i < FDIM * FDIM; i += 256) {
    int k = i >> 6;          // row of W (coalesced global read)
    int n = i & (FDIM - 1);  // col of W
    Wt[n * WT_STRIDE + k] = W[i];
  }
  __syncthreads();

  int wave = t >> 5;                 // 8 waves per block (wave32)
  int lane = t & 31;
  int tile = blockIdx.x * 8 + wave;  // 16-row tile index
  int row0 = tile * 16;
  if (row0 >= nrows) return;         // wave-uniform: EXEC stays all-1s

  int m    = lane & 15;              // row (A) / col (B,C/D) within tile
  int koff = (lane >> 4) * 2;        // 0 for lanes 0-15, 2 for lanes 16-31

  const float* xrow = X + (size_t)(row0 + m) * FDIM + koff;
  const float* wt0  = Wt + (m +  0) * WT_STRIDE + koff;
  const float* wt1  = Wt + (m + 16) * WT_STRIDE + koff;
  const float* wt2  = Wt + (m + 32) * WT_STRIDE + koff;
  const float* wt3  = Wt + (m + 48) * WT_STRIDE + koff;

  v8f acc0 = {}, acc1 = {}, acc2 = {}, acc3 = {};

#pragma unroll
  for (int k0 = 0; k0 < FDIM; k0 += 4) {
    v2f a  = *(const v2f*)(xrow + k0);   // global_load_b64
    v2f b0 = *(const v2f*)(wt0 + k0);    // ds_load_b64, aligned (k0+koff even)
    v2f b1 = *(const v2f*)(wt1 + k0);
    v2f b2 = *(const v2f*)(wt2 + k0);
    v2f b3 = *(const v2f*)(wt3 + k0);
    acc0 = __builtin_amdgcn_wmma_f32_16x16x4_f32(false, a, false, b0, (short)0, acc0, false, false);
    acc1 = __builtin_amdgcn_wmma_f32_16x16x4_f32(false, a, false, b1, (short)0, acc1, false, false);
    acc2 = __builtin_amdgcn_wmma_f32_16x16x4_f32(false, a, false, b2, (short)0, acc2, false, false);
    acc3 = __builtin_amdgcn_wmma_f32_16x16x4_f32(false, a, false, b3, (short)0, acc3, false, false);
  }

  // store C/D: VGPR r -> row (crow + r), col = ccol (+ 16*j per accumulator)
  int crow = (lane < 16) ? 0 : 8;
  int ccol = lane & 15;
  float* yb = Y + (size_t)(row0 + crow) * FDIM + ccol;
#pragma unroll
  for (int r = 0; r < 8; ++r) {
    float* yr = yb + (size_t)r * FDIM;
    yr[0]  = acc0[r];
    yr[16] = acc1[r];
    yr[32] = acc2[r];
    yr[48] = acc3[r];
  }
}

// ---------------- edge scatter: out[d] += dis[s]*dis[d]*xw[s] ----------------
// One wave per edge; lane f handles features {2f, 2f+1} (coalesced b64 gather).
__global__ void gcn_scatter(const int* __restrict__ src,
                            const int* __restrict__ dst,
                            const float* __restrict__ dis,
                            const float* __restrict__ xw,
                            float* __restrict__ out, int E) {
  int gwave  = (blockIdx.x * blockDim.x + threadIdx.x) >> 5;
  int lane   = threadIdx.x & 31;
  int nwaves = (gridDim.x * blockDim.x) >> 5;
  for (int e = gwave; e < E; e += nwaves) {
    int s = src[e];
    int d = dst[e];
    float norm = dis[s] * dis[d];
    const float2 v = *(const float2*)(xw + (size_t)s * FDIM + lane * 2);
    float* o = out + (size_t)d * FDIM + lane * 2;
    atomicAdd(o,     norm * v.x);
    atomicAdd(o + 1, norm * v.y);
  }
}

// ---------------- host-side launch ----------------

extern "C" void kernel_launch(void* const* d_in, const int* in_sizes, int n_in,
                              void* d_out, int out_size, void* d_ws, size_t ws_size,
                              hipStream_t stream) {
  const float* x  = (const float*)d_in[0];
  const int*   ei = (const int*)d_in[1];   // [2, E] int32 (JAX default x64-off)
  const float* W1 = (const float*)d_in[2];
  const float* b1 = (const float*)d_in[3];
  const float* W2 = (const float*)d_in[4];
  const float* b2 = (const float*)d_in[5];

  const int N = in_sizes[0] / FDIM;        // 100000
  const int E = in_sizes[1] / 2;           // 1250000
  const int* src = ei;
  const int* dst = ei + E;

  float* ws   = (float*)d_ws;
  float* dis  = ws;                                        // N floats
  float* buf1 = ws + (((size_t)N + 255) & ~(size_t)255);   // N*64: xw1 then xw2
  float* buf2 = buf1 + (size_t)N * FDIM;                   // N*64: agg1 -> h
  float* out  = (float*)d_out;

  const int nfTotal  = N * FDIM;
  const int gemmBlks = (N / 16 + 7) / 8;                   // 8 waves (tiles)/block
  const int elemBlks = (nfTotal + 255) / 256;
  const int nodeBlks = (N + 255) / 256;

  // --- degrees / normalization ---
  gcn_fill_f32<<<nodeBlks, 256, 0, stream>>>(dis, 1.0f, N);   // self-loop
  gcn_degree_accum<<<2048, 256, 0, stream>>>(dst, dis, E);
  gcn_rsqrt_inplace<<<nodeBlks, 256, 0, stream>>>(dis, N);

  // --- layer 1 ---
  gcn_gemm_wmma<<<gemmBlks, 256, 0, stream>>>(x, W1, buf1, N);
  hipMemsetAsync(buf2, 0, (size_t)nfTotal * sizeof(float), stream);
  gcn_scatter<<<4096, 256, 0, stream>>>(src, dst, dis, buf1, buf2, E);
  gcn_finalize<<<elemBlks, 256, 0, stream>>>(buf2, buf1, dis, b1, nfTotal, 1);

  // --- layer 2 ---
  gcn_gemm_wmma<<<gemmBlks, 256, 0, stream>>>(buf2, W2, buf1, N);
  hipMemsetAsync(out, 0, (size_t)nfTotal * sizeof(float), stream);
  gcn_scatter<<<4096, 256, 0, stream>>>(src, dst, dis, buf1, out, E);
  gcn_finalize<<<elemBlks, 256, 0, stream>>>(out, buf1, dis, b2, nfTotal, 0);
}